// SAGELayer_88974542504019
// MI455X (gfx1250) — compile-verified
//
#include <hip/hip_runtime.h>
#include <hip/hip_bf16.h>
#include <stdint.h>

typedef __attribute__((ext_vector_type(2))) float v2f;
typedef __attribute__((ext_vector_type(8))) float v8f;
typedef int v4i __attribute__((vector_size(16)));
typedef __attribute__((address_space(1))) v4i gv4i;   // global int4
typedef __attribute__((address_space(3))) v4i lv4i;   // LDS int4

#define IN_CH 128
#define WPAD  132                 // LDS row stride (floats): 528B, 16B-aligned, conflict-free
#define BN_EPS 1e-5f

#if defined(__has_builtin)
#  if __has_builtin(__builtin_amdgcn_global_load_async_to_lds_b128) && \
      __has_builtin(__builtin_amdgcn_s_wait_asynccnt)
#    define USE_ASYNC_LDS 1
#  endif
#endif
#ifndef USE_ASYNC_LDS
#  define USE_ASYNC_LDS 0
#endif

// ---------------- zero scratch (agg + deg + stats) ----------------
__global__ void sage_zero_kernel(float4* __restrict__ p, long long n4) {
    long long i = (long long)blockIdx.x * blockDim.x + threadIdx.x;
    if (i < n4) p[i] = make_float4(0.f, 0.f, 0.f, 0.f);
}

// ---------------- edge scatter: agg[dst] += x[src]; deg[dst] += 1 ----------------
// one wave32 per edge, 4 channels per lane (float4 gather + 4 f32 atomics)
__global__ void sage_scatter_kernel(const float* __restrict__ x,
                                    const int* __restrict__ ei,
                                    float* __restrict__ agg,
                                    float* __restrict__ deg,
                                    int n_edges) {
    long long gtid = (long long)blockIdx.x * blockDim.x + threadIdx.x;
    int e    = (int)(gtid >> 5);
    int lane = (int)(gtid & 31);
    if (e >= n_edges) return;
    int s = ei[e];              // src row of edge_index
    int d = ei[n_edges + e];    // dst row of edge_index
    int c = lane << 2;
    float4 v = *(const float4*)(x + (long long)s * IN_CH + c);
    float* a = agg + (long long)d * IN_CH + c;
    __hip_atomic_fetch_add(a + 0, v.x, __ATOMIC_RELAXED, __HIP_MEMORY_SCOPE_AGENT);
    __hip_atomic_fetch_add(a + 1, v.y, __ATOMIC_RELAXED, __HIP_MEMORY_SCOPE_AGENT);
    __hip_atomic_fetch_add(a + 2, v.z, __ATOMIC_RELAXED, __HIP_MEMORY_SCOPE_AGENT);
    __hip_atomic_fetch_add(a + 3, v.w, __ATOMIC_RELAXED, __HIP_MEMORY_SCOPE_AGENT);
    if (lane == 0)
        __hip_atomic_fetch_add(deg + d, 1.0f, __ATOMIC_RELAXED, __HIP_MEMORY_SCOPE_AGENT);
}

// ---------------- fused SAGE GEMM + BN-stat reduction ----------------
// h = (agg/deg) @ Wl^T + b_l + x @ Wr^T  via V_WMMA_F32_16X16X4_F32.
// Weights staged once per block into LDS (async-to-LDS when available),
// shared by 8 waves; one wave per 16-node tile.
__global__ void __launch_bounds__(256)
sage_gemm_kernel(const float* __restrict__ x, const float* __restrict__ agg,
                 const float* __restrict__ deg,
                 const float* __restrict__ Wl, const float* __restrict__ bl,
                 const float* __restrict__ Wr,
                 float* __restrict__ h, float* __restrict__ stats, int n_tiles) {
    extern __shared__ float smem[];
    float* s_wl  = smem;                       // [128][WPAD]
    float* s_wr  = smem + IN_CH * WPAD;        // [128][WPAD]
    float* s_sum = s_wr + IN_CH * WPAD;        // [128]
    float* s_sq  = s_sum + IN_CH;              // [128]

    int tid = threadIdx.x;
    if (tid < IN_CH) { s_sum[tid] = 0.f; s_sq[tid] = 0.f; }

    // cooperative weight staging: 4096 16B chunks per matrix, 16 per thread
#pragma unroll
    for (int it = 0; it < 16; ++it) {
        int idx = it * 256 + tid;
        int r   = idx >> 5;                    // weight row 0..127
        int cc  = (idx & 31) << 2;             // float offset in row
        long long goff = (long long)r * IN_CH + cc;
        int       loff = r * WPAD + cc;
#if USE_ASYNC_LDS
        __builtin_amdgcn_global_load_async_to_lds_b128(
            (gv4i*)(Wl + goff), (lv4i*)(s_wl + loff), 0, 0);
        __builtin_amdgcn_global_load_async_to_lds_b128(
            (gv4i*)(Wr + goff), (lv4i*)(s_wr + loff), 0, 0);
#else
        *(float4*)(s_wl + loff) = *(const float4*)(Wl + goff);
        *(float4*)(s_wr + loff) = *(const float4*)(Wr + goff);
#endif
    }
#if USE_ASYNC_LDS
    __builtin_amdgcn_s_wait_asynccnt(0);
#endif
    __syncthreads();

    int wave = tid >> 5;
    int lane = tid & 31;
    int tile = blockIdx.x * 8 + wave;
    if (tile < n_tiles) {
        int m  = lane & 15;          // A: row within tile / B: output channel within 16-col tile
        int hi = lane >> 4;          // lane half
        int kk = hi << 1;            // K sub-offset 0 or 2 (ISA 16x4 f32 A layout)
        long long row = (long long)tile * 16 + m;
        float invd = 1.0f / fmaxf(deg[row], 1.0f);

        // preload A fragments for all 32 K-steps (x row and degree-normalized agg row)
        v2f ax[32], ag[32];
        const float* xr = x   + row * IN_CH + kk;
        const float* ar = agg + row * IN_CH + kk;
#pragma unroll
        for (int j = 0; j < 32; ++j) {
            ax[j] = *(const v2f*)(xr + j * 4);
            v2f t = *(const v2f*)(ar + j * 4);
            ag[j] = t * invd;
        }

        for (int ot = 0; ot < 8; ++ot) {            // 8 output-channel tiles of 16
            const float* wlr = s_wl + (ot * 16 + m) * WPAD + kk;  // B frag: W[o][k] from LDS
            const float* wrr = s_wr + (ot * 16 + m) * WPAD + kk;
            v8f acc = {};
#pragma unroll
            for (int j = 0; j < 32; ++j) {          // K = 128 in steps of 4
                v2f bL = *(const v2f*)(wlr + j * 4);
                v2f bR = *(const v2f*)(wrr + j * 4);
                acc = __builtin_amdgcn_wmma_f32_16x16x4_f32(
                          false, ag[j], false, bL, (short)0, acc, false, false);
                acc = __builtin_amdgcn_wmma_f32_16x16x4_f32(
                          false, ax[j], false, bR, (short)0, acc, false, false);
            }
            float bias = bl[ot * 16 + m];
            float lsum = 0.f, lsq = 0.f;
#pragma unroll
            for (int r = 0; r < 8; ++r) {           // C/D: row = r + 8*hi, col = m (ISA layout)
                float v = acc[r] + bias;
                int rl = r + (hi << 3);
                h[((long long)tile * 16 + rl) * IN_CH + ot * 16 + m] = v;
                lsum += v;
                lsq  += v * v;
            }
            __hip_atomic_fetch_add(&s_sum[ot * 16 + m], lsum,
                                   __ATOMIC_RELAXED, __HIP_MEMORY_SCOPE_WORKGROUP);
            __hip_atomic_fetch_add(&s_sq[ot * 16 + m], lsq,
                                   __ATOMIC_RELAXED, __HIP_MEMORY_SCOPE_WORKGROUP);
        }
    }
    __syncthreads();
    if (tid < IN_CH) {
        __hip_atomic_fetch_add(stats + tid,         s_sum[tid],
                               __ATOMIC_RELAXED, __HIP_MEMORY_SCOPE_AGENT);
        __hip_atomic_fetch_add(stats + IN_CH + tid, s_sq[tid],
                               __ATOMIC_RELAXED, __HIP_MEMORY_SCOPE_AGENT);
    }
}

// ---------------- BN finalize: per-channel scale/shift ----------------
__global__ void sage_bnfin_kernel(const float* __restrict__ stats,
                                  const float* __restrict__ gamma,
                                  const float* __restrict__ beta,
                                  float* __restrict__ ss, float inv_n) {
    int c = threadIdx.x;
    float mu  = stats[c] * inv_n;
    float var = stats[IN_CH + c] * inv_n - mu * mu;
    float sc  = gamma[c] * rsqrtf(var + BN_EPS);
    ss[c]         = sc;
    ss[IN_CH + c] = beta[c] - mu * sc;
}

// ---------------- apply BN + ReLU in place on h ----------------
__global__ void sage_bnrelu_kernel(float4* __restrict__ h,
                                   const float* __restrict__ ss, long long n4) {
    long long i = (long long)blockIdx.x * blockDim.x + threadIdx.x;
    if (i >= n4) return;
    int c = (int)((i & 31) << 2);   // (i*4) % 128
    float4 v = h[i];
    v.x = fmaxf(v.x * ss[c + 0] + ss[IN_CH + c + 0], 0.f);
    v.y = fmaxf(v.y * ss[c + 1] + ss[IN_CH + c + 1], 0.f);
    v.z = fmaxf(v.z * ss[c + 2] + ss[IN_CH + c + 2], 0.f);
    v.w = fmaxf(v.w * ss[c + 3] + ss[IN_CH + c + 3], 0.f);
    h[i] = v;
}

extern "C" void kernel_launch(void* const* d_in, const int* in_sizes, int n_in,
                              void* d_out, int out_size, void* d_ws, size_t ws_size,
                              hipStream_t stream) {
    const float* x     = (const float*)d_in[0];
    const int*   ei    = (const int*)d_in[1];
    const float* Wl    = (const float*)d_in[2];
    const float* bl    = (const float*)d_in[3];
    const float* Wr    = (const float*)d_in[4];
    const float* gamma = (const float*)d_in[5];
    const float* beta  = (const float*)d_in[6];
    float*       out   = (float*)d_out;

    int n       = in_sizes[0] / IN_CH;   // 50000
    int n_edges = in_sizes[1] / 2;       // 800000
    int n_tiles = n / 16;                // 3125 (exact)

    // workspace layout: [agg n*128][deg n][stats 256][scale/shift 256]
    float* ws    = (float*)d_ws;
    float* agg   = ws;
    float* deg   = ws + (long long)n * IN_CH;
    float* stats = deg + n;
    float* ss    = stats + 2 * IN_CH;

    long long nz  = (long long)n * IN_CH + n + 2 * IN_CH;
    long long nz4 = (nz + 3) / 4;
    sage_zero_kernel<<<(unsigned)((nz4 + 255) / 256), 256, 0, stream>>>((float4*)ws, nz4);

    long long sthreads = (long long)n_edges * 32;
    sage_scatter_kernel<<<(unsigned)((sthreads + 255) / 256), 256, 0, stream>>>(
        x, ei, agg, deg, n_edges);

    size_t shbytes = (size_t)(2 * IN_CH * WPAD + 2 * IN_CH) * sizeof(float);
    sage_gemm_kernel<<<(unsigned)((n_tiles + 7) / 8), 256, shbytes, stream>>>(
        x, agg, deg, Wl, bl, Wr, out, stats, n_tiles);

    sage_bnfin_kernel<<<1, IN_CH, 0, stream>>>(stats, gamma, beta, ss, 1.0f / (float)n);

    long long n4 = (long long)n * IN_CH / 4;
    sage_bnrelu_kernel<<<(unsigned)((n4 + 255) / 256), 256, 0, stream>>>(
        (float4*)out, ss, n4);
}